// CausalWanSelfAttention_45904610460041
// MI455X (gfx1250) — compile-verified
//
#include <hip/hip_runtime.h>

#define DIM 1536
#define NHEADS 12
#define HDIM 128
#define S_LEN 1560
#define KV_LEN 4680
#define START_FRAME 3
#define WIDTH 52

typedef __attribute__((ext_vector_type(2))) float v2f;
typedef __attribute__((ext_vector_type(8))) float v8f;
typedef __attribute__((ext_vector_type(4))) int v4i;

// pointer-to-AS(1)-int4 and pointer-to-AS(3)-int4, as the async builtin expects
typedef __attribute__((address_space(1))) v4i* gv4p;
typedef __attribute__((address_space(3))) v4i* lv4p;

#if defined(__HIP_DEVICE_COMPILE__) &&                                         \
    __has_builtin(__builtin_amdgcn_global_load_async_to_lds_b128) &&           \
    __has_builtin(__builtin_amdgcn_s_wait_asynccnt)
#define USE_ASYNC_LDS 1
#else
#define USE_ASYNC_LDS 0
#endif

// Copy 16B global -> LDS. Async path uses the CDNA5 ASYNCcnt-tracked DMA-style
// copy (no VGPR round trip); fallback is a plain load+store.
__device__ __forceinline__ void cp16_to_lds(float* lds, const float* g) {
#if USE_ASYNC_LDS
  __builtin_amdgcn_global_load_async_to_lds_b128((gv4p)(g), (lv4p)(lds), 0, 0);
#else
  *(float4*)lds = *(const float4*)g;
#endif
}

#if USE_ASYNC_LDS
#define WAIT_ASYNC(n) __builtin_amdgcn_s_wait_asynccnt(n)
#else
#define WAIT_ASYNC(n)
#endif

__device__ __forceinline__ v8f wmma_f32_16x16x4(v2f a, v2f b, v8f c) {
  // V_WMMA_F32_16X16X4_F32: D = A(16x4) * B(4x16) + C(16x16), fp32 throughout
  return __builtin_amdgcn_wmma_f32_16x16x4_f32(false, a, false, b, (short)0, c,
                                               false, false);
}

// ---------------------------------------------------------------------------
// GEMM: Y[M x 1536] = X[M x 1536] @ W[1536 x 1536] + bias
// block 256 thr = 8 waves; tile 64 rows x 128 cols; K staged via LDS in 16s,
// double-buffered with async global->LDS copies.
// wave w handles rows (w&3)*16, cols (w>>2)*64 (4 accumulator 16x16 tiles).
// ---------------------------------------------------------------------------
__global__ __launch_bounds__(256) void gemm_bias_kernel(
    const float* __restrict__ X, const float* __restrict__ W,
    const float* __restrict__ bias, float* __restrict__ Y, int M) {
  __shared__ float Xs[2][64][20];  // padded stride to spread banks
  __shared__ float Ws[2][16][128];
  const int tid  = threadIdx.x;
  const int wave = tid >> 5;
  const int lane = tid & 31;
  const int lm   = lane & 15;
  const int lh   = lane >> 4;
  const int m0   = blockIdx.y * 64;
  const int n0   = blockIdx.x * 128;
  const int wrow = (wave & 3) * 16;
  const int wcol = (wave >> 2) * 64;

  v8f acc[4] = {};

  // 3 async copies per thread per stage (1 for X, 2 for W)
  auto stage = [&](int k0, int b) {
    {
      const int r = tid >> 2;
      const int c = (tid & 3) * 4;
      int gr = m0 + r;
      if (gr >= M) gr = M - 1;  // clamp; stores are guarded later
      cp16_to_lds(&Xs[b][r][c], X + (size_t)gr * DIM + k0 + c);
    }
    {
      const int r = tid >> 4;
      const int c = (tid & 15) * 8;
      const float* src = W + (size_t)(k0 + r) * DIM + n0 + c;
      cp16_to_lds(&Ws[b][r][c], src);
      cp16_to_lds(&Ws[b][r][c + 4], src + 4);
    }
  };

  stage(0, 0);
  const int KSTEPS = DIM / 16;  // 96
  for (int ks = 0; ks < KSTEPS; ++ks) {
    __syncthreads();  // (A) all waves done with buffer (ks+1)&1
    if (ks + 1 < KSTEPS) {
      stage((ks + 1) * 16, (ks + 1) & 1);
      WAIT_ASYNC(8);  // >= 3: this stage's 3 may stay outstanding; 8 does not force prev (3) to land
    } else {
      WAIT_ASYNC(0);
    }
    __syncthreads();  // (B) tile ks visible to all waves
    const int b = ks & 1;
#pragma unroll
    for (int kk = 0; kk < 16; kk += 4) {
      v2f a;
      a.x = Xs[b][wrow + lm][kk + 2 * lh];
      a.y = Xs[b][wrow + lm][kk + 2 * lh + 1];
#pragma unroll
      for (int c = 0; c < 4; ++c) {
        v2f bb;
        const int col = wcol + c * 16 + lm;
        bb.x = Ws[b][kk + 2 * lh][col];
        bb.y = Ws[b][kk + 2 * lh + 1][col];
        acc[c] = wmma_f32_16x16x4(a, bb, acc[c]);
      }
    }
  }

#pragma unroll
  for (int c = 0; c < 4; ++c) {
    const int coln = n0 + wcol + c * 16 + lm;
    const float bv = bias[coln];
#pragma unroll
    for (int r = 0; r < 8; ++r) {
      const int row = m0 + wrow + r + 8 * lh;  // C layout: M = r + 8*(lane/16)
      if (row < M) Y[(size_t)row * DIM + coln] = acc[c][r] + bv;
    }
  }
}

// ---------------------------------------------------------------------------
// In-place RMSNorm (over all 1536 channels) * g, then RoPE per head.
// One block per sequence row. Channel pair p of a head: p<22 -> frame freq
// (row START_FRAME), 22<=p<43 -> row h=s/52, else row w=s%52.
// ---------------------------------------------------------------------------
__global__ __launch_bounds__(256) void rms_rope_kernel(
    float* __restrict__ Y, const float* __restrict__ g,
    const float* __restrict__ fr, const float* __restrict__ fi) {
  const int row = blockIdx.x;
  float* y = Y + (size_t)row * DIM;
  const int tid = threadIdx.x;

  float ss = 0.f;
  for (int c = tid; c < DIM; c += 256) {
    const float v = y[c];
    ss += v * v;
  }
  __shared__ float red[256];
  red[tid] = ss;
  __syncthreads();
  for (int off = 128; off > 0; off >>= 1) {
    if (tid < off) red[tid] += red[tid + off];
    __syncthreads();
  }
  const float inv = rsqrtf(red[0] * (1.0f / DIM) + 1e-6f);

  const int h = row / WIDTH;
  const int w = row % WIDTH;
  for (int pg = tid; pg < NHEADS * 64; pg += 256) {
    const int p = pg & 63;
    const int c = (pg >> 6) * HDIM + 2 * p;
    const float x0 = y[c] * inv * g[c];
    const float x1 = y[c + 1] * inv * g[c + 1];
    int fidx;
    if (p < 22)      fidx = START_FRAME * 64 + p;
    else if (p < 43) fidx = h * 64 + p;
    else             fidx = w * 64 + p;
    const float cr = fr[fidx];
    const float ci = fi[fidx];
    y[c]     = x0 * cr - x1 * ci;
    y[c + 1] = x0 * ci + x1 * cr;
  }
}

// ---------------------------------------------------------------------------
// Flash attention, one head per blockIdx.x, 4 query tiles (4 waves) per block.
// Keys gathered: j<1560 -> cacheK[j]; 1560<=j<3120 -> cacheK[j+1560];
// j>=3120 -> Knew[j-3120] (same for V). Online softmax, fp32 WMMA.
// K/V tiles are double-buffered through LDS with async copies; Q A-fragments
// live in registers (loaded once). Masked tail keys load clamped (finite)
// data; softmax forces their probabilities to exactly 0.
// ---------------------------------------------------------------------------
__global__ __launch_bounds__(128) void attn_kernel(
    const float* __restrict__ Q, const float* __restrict__ Knew,
    const float* __restrict__ Vnew, const float* __restrict__ cacheK,
    const float* __restrict__ cacheV, float* __restrict__ O) {
  constexpr int LSTR = 132;
  __shared__ float Klds[2][16][LSTR];
  __shared__ float Vlds[2][16][LSTR];
  __shared__ float Slds[4][16][18];
  __shared__ float Alds[4][16];
  __shared__ float Llds[4][16];

  const int head = blockIdx.x;
  const int qg   = blockIdx.y;
  const int tid  = threadIdx.x;
  const int wave = tid >> 5;
  const int lane = tid & 31;
  const int lm   = lane & 15;
  const int lh   = lane >> 4;

  int qt = qg * 4 + wave;
  const int QT_MAX = (S_LEN + 15) / 16 - 1;  // 97
  if (qt > QT_MAX) qt = QT_MAX;              // duplicated work, stores identical

  // Q A-fragments in registers: lane holds row lane%16, col pairs 4i+2*(lane/16)
  v2f qf[32];
  {
    int grow = qt * 16 + lm;
    if (grow >= S_LEN) grow = 0;  // finite filler; rows never stored
    const float* src = Q + (size_t)grow * DIM + head * HDIM;
#pragma unroll
    for (int i = 0; i < 32; ++i)
      qf[i] = *(const v2f*)(src + 4 * i + 2 * lh);
  }

  // 8 async copies per thread per staged tile (4 K + 4 V)
  auto stage_kv = [&](int t, int b) {
    const int r = tid >> 3;
    const int cbase = (tid & 7) * 16;
    int key = t * 16 + r;
    if (key >= KV_LEN) key = 0;  // clamp; these columns get p == 0 in softmax
    int tok;
    const float *kp, *vp;
    if (key < 1560) {
      tok = key; kp = cacheK; vp = cacheV;                // sink
    } else if (key < 3120) {
      tok = key + 1560; kp = cacheK; vp = cacheV;         // rolled window
    } else {
      tok = key - 3120; kp = Knew; vp = Vnew;             // fresh tokens
    }
    const size_t off = (size_t)tok * DIM + head * HDIM + cbase;
#pragma unroll
    for (int c = 0; c < 16; c += 4) {
      cp16_to_lds(&Klds[b][r][cbase + c], kp + off + c);
      cp16_to_lds(&Vlds[b][r][cbase + c], vp + off + c);
    }
  };

  v8f oacc[8] = {};
  float mrow = -1e30f;
  float lrow = 0.f;
  const float scale = 0.088388347648318447f;  // 1/sqrt(128)

  const int KV_TILES = (KV_LEN + 15) / 16;  // 293
  stage_kv(0, 0);
  for (int t = 0; t < KV_TILES; ++t) {
    __syncthreads();  // (A) all waves done computing on buffer (t+1)&1
    if (t + 1 < KV_TILES) {
      stage_kv(t + 1, (t + 1) & 1);
      WAIT_ASYNC(8);  // in-order: <=8 outstanding => tile t landed
    } else {
      WAIT_ASYNC(0);
    }
    __syncthreads();  // (B) tile t visible to all waves
    const int b = t & 1;

    // S = Q * K^T : 32 WMMA over the 128-dim
    v8f s = {};
#pragma unroll
    for (int dd = 0; dd < HDIM; dd += 4) {
      v2f bb;
      bb.x = Klds[b][lm][dd + 2 * lh];
      bb.y = Klds[b][lm][dd + 2 * lh + 1];
      s = wmma_f32_16x16x4(qf[dd >> 2], bb, s);
    }
#pragma unroll
    for (int r = 0; r < 8; ++r)
      Slds[wave][r + 8 * lh][lm] = s[r] * scale;
    __syncthreads();

    // online softmax row update (lanes 0..15, one row each)
    if (lh == 0) {
      const int r = lm;
      const int kb = t * 16;
      float sv[16];
      float tmax = -1e30f;
#pragma unroll
      for (int j = 0; j < 16; ++j) {
        const float v = (kb + j < KV_LEN) ? Slds[wave][r][j] : -1e30f;
        sv[j] = v;
        tmax = fmaxf(tmax, v);
      }
      const float mnew = fmaxf(mrow, tmax);
      const float alpha = __expf(mrow - mnew);
      float psum = 0.f;
#pragma unroll
      for (int j = 0; j < 16; ++j) {
        const float p = (kb + j < KV_LEN) ? __expf(sv[j] - mnew) : 0.f;
        Slds[wave][r][j] = p;
        psum += p;
      }
      lrow = lrow * alpha + psum;
      mrow = mnew;
      Alds[wave][r] = alpha;
    }
    __syncthreads();

    // rescale running O by alpha, then O += P * V (32 WMMA)
    float ar[8];
#pragma unroll
    for (int r = 0; r < 8; ++r) ar[r] = Alds[wave][r + 8 * lh];
#pragma unroll
    for (int c = 0; c < 8; ++c)
#pragma unroll
      for (int r = 0; r < 8; ++r) oacc[c][r] *= ar[r];

#pragma unroll
    for (int c = 0; c < 8; ++c) {
#pragma unroll
      for (int kk = 0; kk < 16; kk += 4) {
        v2f a, bb;
        a.x = Slds[wave][lm][kk + 2 * lh];
        a.y = Slds[wave][lm][kk + 2 * lh + 1];
        const int col = c * 16 + lm;
        bb.x = Vlds[b][kk + 2 * lh][col];
        bb.y = Vlds[b][kk + 2 * lh + 1][col];
        oacc[c] = wmma_f32_16x16x4(a, bb, oacc[c]);
      }
    }
  }

  __syncthreads();
  if (lh == 0) Llds[wave][lm] = lrow;
  __syncthreads();
  float li[8];
#pragma unroll
  for (int r = 0; r < 8; ++r) li[r] = 1.f / Llds[wave][r + 8 * lh];

#pragma unroll
  for (int c = 0; c < 8; ++c) {
    const int coln = head * HDIM + c * 16 + lm;
#pragma unroll
    for (int r = 0; r < 8; ++r) {
      const int row = qt * 16 + r + 8 * lh;
      if (row < S_LEN) O[(size_t)row * DIM + coln] = oacc[c][r] * li[r];
    }
  }
}

// ---------------------------------------------------------------------------
extern "C" void kernel_launch(void* const* d_in, const int* in_sizes, int n_in,
                              void* d_out, int out_size, void* d_ws,
                              size_t ws_size, hipStream_t stream) {
  (void)in_sizes; (void)n_in; (void)out_size; (void)ws_size;
  const float* x       = (const float*)d_in[0];
  const float* cache_k = (const float*)d_in[1];
  const float* cache_v = (const float*)d_in[2];
  const float* fr      = (const float*)d_in[3];
  const float* fi      = (const float*)d_in[4];
  const float* wq      = (const float*)d_in[5];
  const float* bq      = (const float*)d_in[6];
  const float* wk      = (const float*)d_in[7];
  const float* bk      = (const float*)d_in[8];
  const float* wv      = (const float*)d_in[9];
  const float* bv      = (const float*)d_in[10];
  const float* wo      = (const float*)d_in[11];
  const float* bo      = (const float*)d_in[12];
  const float* gq      = (const float*)d_in[13];
  const float* gk      = (const float*)d_in[14];
  float* out = (float*)d_out;

  const size_t plane = (size_t)S_LEN * DIM;
  float* Qbuf = (float*)d_ws;
  float* Knew = Qbuf + plane;
  float* Vnew = Knew + plane;
  float* Oat  = Vnew + plane;

  const dim3 ggrid(DIM / 128, (S_LEN + 63) / 64);
  gemm_bias_kernel<<<ggrid, 256, 0, stream>>>(x, wq, bq, Qbuf, S_LEN);
  gemm_bias_kernel<<<ggrid, 256, 0, stream>>>(x, wk, bk, Knew, S_LEN);
  gemm_bias_kernel<<<ggrid, 256, 0, stream>>>(x, wv, bv, Vnew, S_LEN);

  rms_rope_kernel<<<S_LEN, 256, 0, stream>>>(Qbuf, gq, fr, fi);
  rms_rope_kernel<<<S_LEN, 256, 0, stream>>>(Knew, gk, fr, fi);

  const int qgroups = ((S_LEN + 15) / 16 + 3) / 4;  // 25
  attn_kernel<<<dim3(NHEADS, qgroups), 128, 0, stream>>>(Qbuf, Knew, Vnew,
                                                         cache_k, cache_v, Oat);

  gemm_bias_kernel<<<ggrid, 256, 0, stream>>>(Oat, wo, bo, out, S_LEN);
}